// NomicRouter_42829413875909
// MI455X (gfx1250) — compile-verified
//
#include <hip/hip_runtime.h>
#include <math.h>

typedef __attribute__((ext_vector_type(2))) float v2f;
typedef __attribute__((ext_vector_type(4))) float v4f;
typedef __attribute__((ext_vector_type(8))) float v8f;

namespace {
constexpr int kN    = 16384;  // tokens
constexpr int kH    = 2048;   // hidden
constexpr int kE    = 8;      // experts
constexpr int kTopK = 2;
constexpr int kKC   = 32;     // K-chunk staged in LDS per iteration
constexpr int kWaves = 8;     // wave32s per block
constexpr int kRowsPerWave  = 16;
constexpr int kRowsPerBlock = kWaves * kRowsPerWave;  // 128
constexpr int kXtStride = 34; // odd-ish stride dodges bank conflicts; 136B rows
                              // keep 8B alignment so A-pairs fuse to ds_load_b64
constexpr int kWPad = 16;     // W tile padded 8 -> 16 experts (zeros), so B
                              // fragment loads need no predication / exec masks
}

__global__ __launch_bounds__(kWaves * 32)
void nomic_router_wmma(const float* __restrict__ x,
                       const float* __restrict__ W,
                       float* __restrict__ out_w,
                       float* __restrict__ out_topw,
                       long long* __restrict__ out_tope)
{
  // All LDS is strictly per-wave: no __syncthreads needed anywhere —
  // same-wave DS ordering (DScnt, in-order) covers every RAW below.
  __shared__ float xt [kWaves][kRowsPerWave][kXtStride];  // ~17.4 KB
  __shared__ float wcw[kWaves][kKC][kWPad];               //  16 KB
  __shared__ float wsm[kWaves][kRowsPerWave][kE];         //   4 KB

  const int tid  = threadIdx.x;
  const int wave = tid >> 5;
  const int lane = tid & 31;
  const int hi   = lane >> 4;   // half-wave: K pair {0,1} vs {2,3}
  const int l    = lane & 15;   // M index for A; N (expert) index for B/C
  const int waveRow0 = blockIdx.x * kRowsPerBlock + wave * kRowsPerWave;

  // Zero the expert-padding columns once; the k-loop never writes them.
  {
    v4f z = {0.f, 0.f, 0.f, 0.f};
    *(v4f*)&wcw[wave][lane][8]  = z;
    *(v4f*)&wcw[wave][lane][12] = z;
  }

  // Per-lane x-tile slot: idx = i*32+lane -> (row = idx>>3, col4 = idx&7);
  // 8 consecutive lanes cover one 128B row segment -> coalesced b128 loads.
  // Keep plain integer offsets (indexing off `x` directly) so address-space
  // inference yields global_load_b128, not flat_load (flat ties up DScnt).
  size_t xoff[4];
  int xrow[4], xcol[4];
#pragma unroll
  for (int i = 0; i < 4; ++i) {
    const int idx = i * 32 + lane;
    xrow[i] = idx >> 3;
    xcol[i] = (idx & 7) << 2;
    xoff[i] = (size_t)(waveRow0 + xrow[i]) * kH + xcol[i];
  }
  // W chunk: lane owns row k = k0 + lane (8 contiguous floats, row-major W).
  const size_t woff = (size_t)lane * kE;

  v8f acc = {0.f, 0.f, 0.f, 0.f, 0.f, 0.f, 0.f, 0.f};

  for (int k0 = 0; k0 < kH; k0 += kKC) {
    // Stage W chunk [32 x 8] -> padded [32 x 16] (cols 8..15 stay zero).
    {
      const float* wp = W + woff + (size_t)k0 * kE;
      const v4f w0 = *(const v4f*)(wp + 0);
      const v4f w1 = *(const v4f*)(wp + 4);
      *(v4f*)&wcw[wave][lane][0] = w0;
      *(v4f*)&wcw[wave][lane][4] = w1;
    }
    // Stage x tile [16 x 32]: issue all four b128 loads first (distinct
    // temporaries -> one s_clause, overlapped latencies), then drain into LDS.
    v4f tv[4];
#pragma unroll
    for (int i = 0; i < 4; ++i) {
      tv[i] = *(const v4f*)(x + xoff[i] + k0);
    }
#pragma unroll
    for (int i = 0; i < 4; ++i) {
      float* dst = &xt[wave][xrow[i]][xcol[i]];
      dst[0] = tv[i].x; dst[1] = tv[i].y; dst[2] = tv[i].z; dst[3] = tv[i].w;
    }
    // Hint the next chunk (global_prefetch_b8); speculative prefetch past the
    // end of x is silently dropped, so no guard branch needed.
    __builtin_prefetch(x + xoff[0] + k0 + kKC, 0, 1);

    // 8 x V_WMMA_F32_16X16X4_F32, full EXEC, no predication in the loop.
#pragma unroll
    for (int kk = 0; kk < kKC; kk += 4) {
      const int ka = kk + 2 * hi;
      v2f a, b;
      a.x = xt[wave][l][ka + 0];     // A: M=l, K=ka/ka+1 (fuses to ds_load_b64)
      a.y = xt[wave][l][ka + 1];
      b.x = wcw[wave][ka + 0][l];    // B: N=l, K=ka/ka+1 (pad cols read zeros)
      b.y = wcw[wave][ka + 1][l];
      acc = __builtin_amdgcn_wmma_f32_16x16x4_f32(false, a, false, b,
                                                  (short)0, acc, false, false);
    }
  }

  // acc[v] = logits[row = v + 8*hi][expert = l] (valid for l < 8).
  // Softmax across experts = across lanes {0..7} / {16..23}: xor masks 1,2,4
  // stay inside each 8-lane group, so the zero-padded lanes never mix in.
  float wrow[8];
#pragma unroll
  for (int v = 0; v < 8; ++v) {
    const float t = acc[v];
    float m = t;
    m = fmaxf(m, __shfl_xor(m, 1));
    m = fmaxf(m, __shfl_xor(m, 2));
    m = fmaxf(m, __shfl_xor(m, 4));
    const float e = expf(t - m);
    float s = e;
    s += __shfl_xor(s, 1);
    s += __shfl_xor(s, 2);
    s += __shfl_xor(s, 4);
    wrow[v] = e / s;
  }

  // Store dense softmax weights straight from the WMMA layout; stage to LDS
  // for the per-row top-2 scan (same-wave DS ordering, no barrier).
  if (l < kE) {
#pragma unroll
    for (int v = 0; v < 8; ++v) {
      const int row = v + 8 * hi;
      out_w[(size_t)(waveRow0 + row) * kE + l] = wrow[v];
      wsm[wave][row][l] = wrow[v];
    }
  }

  // Lanes 0..15: one row each; strict '>' matches top_k's low-index tie-break.
  if (hi == 0) {
    const int row = l;
    float b0 = -1.f, b1 = -1.f;
    int i0 = 0, i1 = 0;
#pragma unroll
    for (int e = 0; e < kE; ++e) {
      const float v = wsm[wave][row][e];
      if (v > b0)      { b1 = b0; i1 = i0; b0 = v; i0 = e; }
      else if (v > b1) { b1 = v; i1 = e; }
    }
    const size_t g = (size_t)(waveRow0 + row);
    out_topw[g * kTopK + 0] = b0;
    out_topw[g * kTopK + 1] = b1;
    out_tope[g * kTopK + 0] = (long long)i0;
    out_tope[g * kTopK + 1] = (long long)i1;
  }
}

extern "C" void kernel_launch(void* const* d_in, const int* in_sizes, int n_in,
                              void* d_out, int out_size, void* d_ws, size_t ws_size,
                              hipStream_t stream) {
  const float* x = (const float*)d_in[0];   // [16384, 2048] f32
  const float* W = (const float*)d_in[1];   // [2048, 8] f32
  // Outputs concatenated in return order: weights [N,8] f32,
  // top_weights [N,2] f32, top_experts [N,2] i64 (offset is 8-byte aligned).
  float*     out_w    = (float*)d_out;
  float*     out_topw = out_w + (size_t)kN * kE;
  long long* out_tope = (long long*)(out_topw + (size_t)kN * kTopK);

  dim3 grid(kN / kRowsPerBlock);   // 128 blocks
  dim3 block(kWaves * 32);         // 256 threads = 8 wave32s
  nomic_router_wmma<<<grid, block, 0, stream>>>(x, W, out_w, out_topw, out_tope);
}